// QuaternionBatchNorm_78091095375898
// MI455X (gfx1250) — compile-verified
//
#include <hip/hip_runtime.h>

typedef float v2f __attribute__((ext_vector_type(2)));
typedef float v8f __attribute__((ext_vector_type(8)));

#define BATCH 16384
#define FDIM  2048
#define NSPLIT 64
#define ROWS_PER_SPLIT (BATCH / NSPLIT)   // 256
#define PART_STRIDE 20                    // 16 S2 + 4 S1 (also reused for 16 M + 4 bias)
#define APPLY_SPLIT 128
#define ROWS3 (BATCH / APPLY_SPLIT)       // 128
#define EPSV 1e-5f

// ---------------------------------------------------------------------------
// Pass 1: per-feature raw sums S1 = sum_b x, S2 = sum_b x x^T via
// V_WMMA_F32_16X16X4_F32. Four features packed block-diagonally:
//   A (16x4, MxK): row m = 4*feat + dim, col k = batch within group of 4
//   A layout (ISA 7.12.2): lane = M (both halves), VGPR v holds K = v + 2*(lane>=16)
//   B gets the identical per-lane data => B = A^T => D += A*A^T (Gram update).
// Diagonal 4x4 blocks of D are the per-feature second-moment matrices.
// ---------------------------------------------------------------------------
__global__ __launch_bounds__(256) void qbn_stats(const float* __restrict__ x,
                                                 float* __restrict__ part) {
  const int lane = threadIdx.x & 31;
  const int wave = threadIdx.x >> 5;
  const int f0   = blockIdx.x * 32 + wave * 4;    // 4 features per wave
  const int l    = lane & 15;                     // element within 16-float group
  const int hi   = lane >> 4;                     // K half: batches {0,1} vs {2,3}
  const size_t rowStride = (size_t)FDIM * 4;      // floats per batch row
  const size_t b0 = (size_t)blockIdx.y * ROWS_PER_SPLIT;

  v8f acc = {};        // 16x16 accumulator (8 VGPRs)
  float s1 = 0.f;

  size_t base = (b0 + 2 * hi) * rowStride + (size_t)f0 * 4 + l;
#pragma unroll 4
  for (int k = 0; k < ROWS_PER_SPLIT; k += 4) {
    float va = x[base];                 // batch b0+k+2*hi
    float vb = x[base + rowStride];     // batch b0+k+2*hi+1
    v2f a; a.x = va; a.y = vb;
    acc = __builtin_amdgcn_wmma_f32_16x16x4_f32(false, a, false, a,
                                                (short)0, acc, false, false);
    s1 += va + vb;
    base += 4 * rowStride;
  }

  // Each lane covered half the batches for its (feature,dim); fold halves.
  s1 += __shfl_xor(s1, 16, 32);

  // D layout (32-bit C/D 16x16): VGPR r -> M = r (lanes 0-15) / r+8 (lanes 16-31),
  // N = lane&15. Useful diagonal blocks:
  //   features 0,1: lanes 0-7   hold S2[h][i][j] in acc[4h+i],   j = lane&3
  //   features 2,3: lanes 24-31 hold S2[h][i][j] in acc[4h+i-8], j = lane&3
  if (lane < 16) {   // S1: lane = 4g+i -> feature f0+g, dim i
    size_t pb = ((size_t)blockIdx.y * FDIM + f0 + (lane >> 2)) * PART_STRIDE;
    part[pb + 16 + (lane & 3)] = s1;
  }
  if (lane < 8) {
    int h = lane >> 2, j = lane & 3;
    size_t pb = ((size_t)blockIdx.y * FDIM + f0 + h) * PART_STRIDE;
#pragma unroll
    for (int i = 0; i < 4; ++i) part[pb + i * 4 + j] = acc[4 * h + i];
  } else if (lane >= 24) {
    int h = (lane - 16) >> 2, j = lane & 3;      // h = 2 or 3
    size_t pb = ((size_t)blockIdx.y * FDIM + f0 + h) * PART_STRIDE;
#pragma unroll
    for (int i = 0; i < 4; ++i) part[pb + i * 4 + j] = acc[4 * (h - 2) + i];
  }
}

// ---------------------------------------------------------------------------
// Pass 2: reduce partials; cov = S2/B - mu mu^T + eps*I; 4x4 Cholesky;
// L_inv by forward substitution; M = gamma_sym * L_inv; bias = beta - M*mu.
// ---------------------------------------------------------------------------
__global__ __launch_bounds__(256) void qbn_finalize(const float* __restrict__ part,
                                                    const float* __restrict__ gamma,
                                                    const float* __restrict__ beta,
                                                    float* __restrict__ coef) {
  const int f = blockIdx.x * 256 + threadIdx.x;
  if (f >= FDIM) return;

  float S2[16], S1[4];
#pragma unroll
  for (int e = 0; e < 16; ++e) S2[e] = 0.f;
#pragma unroll
  for (int d = 0; d < 4; ++d) S1[d] = 0.f;
  for (int s = 0; s < NSPLIT; ++s) {
    const float* p = part + ((size_t)s * FDIM + f) * PART_STRIDE;
#pragma unroll
    for (int e = 0; e < 16; ++e) S2[e] += p[e];
#pragma unroll
    for (int d = 0; d < 4; ++d) S1[d] += p[16 + d];
  }

  const float invB = 1.0f / (float)BATCH;
  float mu[4];
#pragma unroll
  for (int d = 0; d < 4; ++d) mu[d] = S1[d] * invB;

  float C[4][4];
#pragma unroll
  for (int i = 0; i < 4; ++i)
#pragma unroll
    for (int j = 0; j < 4; ++j)
      C[i][j] = S2[i * 4 + j] * invB - mu[i] * mu[j];
#pragma unroll
  for (int i = 0; i < 4; ++i) C[i][i] += EPSV;

  // Cholesky (lower)
  float L00 = sqrtf(C[0][0]);               float iL00 = 1.f / L00;
  float L10 = C[1][0] * iL00, L20 = C[2][0] * iL00, L30 = C[3][0] * iL00;
  float L11 = sqrtf(C[1][1] - L10 * L10);   float iL11 = 1.f / L11;
  float L21 = (C[2][1] - L20 * L10) * iL11;
  float L31 = (C[3][1] - L30 * L10) * iL11;
  float L22 = sqrtf(C[2][2] - L20 * L20 - L21 * L21); float iL22 = 1.f / L22;
  float L32 = (C[3][2] - L30 * L20 - L31 * L21) * iL22;
  float L33 = sqrtf(C[3][3] - L30 * L30 - L31 * L31 - L32 * L32);
  float iL33 = 1.f / L33;

  // Triangular inverse
  float Li10 = -L10 * iL00 * iL11;
  float Li20 = -(L20 * iL00 + L21 * Li10) * iL22;
  float Li21 = -L21 * iL11 * iL22;
  float Li30 = -(L30 * iL00 + L31 * Li10 + L32 * Li20) * iL33;
  float Li31 = -(L31 * iL11 + L32 * Li21) * iL33;
  float Li32 = -L32 * iL22 * iL33;
  float Linv[4][4] = {{iL00, 0.f, 0.f, 0.f},
                      {Li10, iL11, 0.f, 0.f},
                      {Li20, Li21, iL22, 0.f},
                      {Li30, Li31, Li32, iL33}};

  // Symmetric gamma from lower-tri flat [10]
  const float* gm = gamma + (size_t)f * 10;
  float G[4][4];
  G[0][0] = gm[0];
  G[1][0] = G[0][1] = gm[1]; G[1][1] = gm[2];
  G[2][0] = G[0][2] = gm[3]; G[2][1] = G[1][2] = gm[4]; G[2][2] = gm[5];
  G[3][0] = G[0][3] = gm[6]; G[3][1] = G[1][3] = gm[7]; G[3][2] = G[2][3] = gm[8];
  G[3][3] = gm[9];

  float* c = coef + (size_t)f * PART_STRIDE;
  float M[4][4];
#pragma unroll
  for (int i = 0; i < 4; ++i) {
#pragma unroll
    for (int j = 0; j < 4; ++j) {
      float acc = 0.f;
#pragma unroll
      for (int k = 0; k < 4; ++k) acc = fmaf(G[i][k], Linv[k][j], acc);
      M[i][j] = acc;
      c[i * 4 + j] = acc;
    }
  }
#pragma unroll
  for (int i = 0; i < 4; ++i) {
    float b = beta[(size_t)f * 4 + i];
#pragma unroll
    for (int j = 0; j < 4; ++j) b -= M[i][j] * mu[j];
    c[16 + i] = b;
  }
}

// ---------------------------------------------------------------------------
// Pass 3: out[b,f,:] = M_f * x[b,f,:] + bias_f. One thread per feature column,
// float4 (b128) loads/stores, 512B contiguous per wave per row, prefetch ahead.
// ---------------------------------------------------------------------------
__global__ __launch_bounds__(256) void qbn_apply(const float* __restrict__ x,
                                                 const float* __restrict__ coef,
                                                 float* __restrict__ out) {
  const int f = blockIdx.x * 256 + threadIdx.x;
  const float* c = coef + (size_t)f * PART_STRIDE;
  float m[16], bv[4];
#pragma unroll
  for (int i = 0; i < 16; ++i) m[i] = c[i];
#pragma unroll
  for (int i = 0; i < 4; ++i) bv[i] = c[16 + i];

  const float4* xv = (const float4*)x;
  float4* ov = (float4*)out;
  size_t idx = (size_t)blockIdx.y * ROWS3 * FDIM + f;
#pragma unroll 4
  for (int r = 0; r < ROWS3; ++r) {
    __builtin_prefetch((const void*)(xv + idx + 8 * (size_t)FDIM), 0, 1);
    float4 v = xv[idx];
    float4 o;
    o.x = fmaf(m[0],  v.x, fmaf(m[1],  v.y, fmaf(m[2],  v.z, fmaf(m[3],  v.w, bv[0]))));
    o.y = fmaf(m[4],  v.x, fmaf(m[5],  v.y, fmaf(m[6],  v.z, fmaf(m[7],  v.w, bv[1]))));
    o.z = fmaf(m[8],  v.x, fmaf(m[9],  v.y, fmaf(m[10], v.z, fmaf(m[11], v.w, bv[2]))));
    o.w = fmaf(m[12], v.x, fmaf(m[13], v.y, fmaf(m[14], v.z, fmaf(m[15], v.w, bv[3]))));
    ov[idx] = o;
    idx += FDIM;
  }
}

extern "C" void kernel_launch(void* const* d_in, const int* in_sizes, int n_in,
                              void* d_out, int out_size, void* d_ws, size_t ws_size,
                              hipStream_t stream) {
  const float* x     = (const float*)d_in[0];
  const float* gamma = (const float*)d_in[1];
  const float* beta  = (const float*)d_in[2];
  float* out  = (float*)d_out;
  float* part = (float*)d_ws;                                    // [NSPLIT][F][20]
  float* coef = part + (size_t)NSPLIT * FDIM * PART_STRIDE;      // [F][20]

  dim3 g1(FDIM / 32, NSPLIT);              // 1 wave = 4 features, 8 waves/block
  qbn_stats<<<g1, 256, 0, stream>>>(x, part);

  qbn_finalize<<<FDIM / 256, 256, 0, stream>>>(part, gamma, beta, coef);

  dim3 g3(FDIM / 256, APPLY_SPLIT);
  qbn_apply<<<g3, 256, 0, stream>>>(x, coef, out);
}